// EMAMomentumEmbedding_90649579749472
// MI455X (gfx1250) — compile-verified
//
#include <hip/hip_runtime.h>
#include <math.h>

typedef __attribute__((ext_vector_type(2))) float v2f;
typedef __attribute__((ext_vector_type(8))) float v8f;

namespace {
constexpr int   kS     = 2048;
constexpr int   kD     = 2048;
constexpr int   kB     = 8;
constexpr float kBeta  = 0.9f;
constexpr int   kTiles = kS / 16;   // 128 time-tiles per sequence
}

// One wave32 owns a 16-wide d-tile of one batch row. Per 16-step time-tile:
//   momentum_tile(16x16) = W(16x16) x V(16x16) + cumA (x) carry
// via four chained V_WMMA_F32_16X16X4_F32 (carry folded into the C operand).
// The loop is software-pipelined: idx gathers run 2 tiles ahead, p/q value
// gathers run 1 tile ahead, so the serial carry chain only pays WMMA latency.
__global__ __launch_bounds__(128, 1) void ema_momentum_wmma(
    const int* __restrict__ idx,
    const float* __restrict__ reset_mask,
    const float* __restrict__ q_emb,
    const float* __restrict__ p_emb,
    float* __restrict__ out)
{
  const int lane   = threadIdx.x & 31;
  const int lane16 = lane & 15;
  const int hi     = lane >> 4;                 // 0: lanes 0-15, 1: lanes 16-31

  const int wave  = blockIdx.x * (blockDim.x >> 5) + (threadIdx.x >> 5);
  const int b     = wave >> 7;                  // 128 d-tiles per batch row
  const int dtile = wave & 127;
  const unsigned d = (unsigned)(dtile * 16 + lane16);   // this lane's column (N)

  const int*   idx_b = idx        + b * kS;
  const float* rm_b  = reset_mask + b * kS;
  // per-lane output base: row block start for this half-wave, column d
  float* out_p = out + (size_t)b * kS * kD + (size_t)(hi * 8) * kD + d;

  // pipeline registers
  int   i1c[8], i0c[8], qic[8];   // idx for tile n+1 (feeds next p/q issue)
  int   i1n[8], i0n[8], qin[8];   // idx for tile n+2 (in flight)
  float p1c[8], p0c[8], qvc[8];   // p/q values for tile n (being computed)
  float p1n[8], p0n[8], qvn[8];   // p/q values for tile n+1 (in flight)

  // ---- prologue: tile 0 idx -> tile 0 values; tile 1 idx -------------------
#pragma unroll
  for (int e = 0; e < 8; ++e) {
    const int tb = 4 * (e >> 1) + 2 * (e & 1) + hi;   // velocity row in tile
    i1c[e] = idx_b[tb];
    i0c[e] = idx_b[(tb > 0) ? (tb - 1) : 0];          // t==0 -> velocity 0
    qic[e] = idx_b[e + 8 * hi];
  }
#pragma unroll
  for (int e = 0; e < 8; ++e) {
    p1c[e] = p_emb[(((unsigned)i1c[e]) << 11) + d];
    p0c[e] = p_emb[(((unsigned)i0c[e]) << 11) + d];
    qvc[e] = q_emb[(((unsigned)qic[e]) << 11) + d];
  }
#pragma unroll
  for (int e = 0; e < 8; ++e) {
    const int tb = 4 * (e >> 1) + 2 * (e & 1) + hi;
    i1c[e] = idx_b[16 + tb];
    i0c[e] = idx_b[15 + tb];
    qic[e] = idx_b[16 + e + 8 * hi];
  }
  float r_cur = rm_b[lane16];     // reset for tile 0, step lane16 (both halves)
  float carry = 0.0f;

  for (int n = 0; n < kTiles; ++n) {
    const int t0   = n * 16;
    const int t0n1 = (n + 1 < kTiles) ? t0 + 16 : t0;   // clamped lookahead
    const int t0n2 = (n + 2 < kTiles) ? t0 + 32 : t0n1;

    // (1) idx gathers for tile n+2 (t0n2 >= 32 here, no t==0 clamp needed)
#pragma unroll
    for (int e = 0; e < 8; ++e) {
      const int tb = 4 * (e >> 1) + 2 * (e & 1) + hi;
      i1n[e] = idx_b[t0n2 + tb];
      i0n[e] = idx_b[t0n2 + tb - 1];
      qin[e] = idx_b[t0n2 + e + 8 * hi];
    }
    // (2) p/q value gathers for tile n+1 (addresses ready from last iteration)
#pragma unroll
    for (int e = 0; e < 8; ++e) {
      p1n[e] = p_emb[(((unsigned)i1c[e]) << 11) + d];
      p0n[e] = p_emb[(((unsigned)i0c[e]) << 11) + d];
      qvn[e] = q_emb[(((unsigned)qic[e]) << 11) + d];
    }
    // (3) reset scalars for tile n+1
    const float r_next = rm_b[t0n1 + lane16];

    // (4) compute tile n ----------------------------------------------------
    const float r   = r_cur;
    const bool  rst = (r > 0.5f);
    const float a   = kBeta * (1.0f - r);             // decay a_t
    // inclusive prefix product of (rst ? 1 : a) (never zero) and last-reset
    // position (-1 if none) over the 16 steps; lanes 16-31 replicate t=0..15
    float P       = rst ? 1.0f : a;
    int   lastRst = rst ? lane16 : -1;
#pragma unroll
    for (int off = 1; off < 16; off <<= 1) {
      const float Pn = __shfl(P, lane - off, 32);
      const int   Ln = __shfl(lastRst, lane - off, 32);
      if (lane16 >= off) {
        P *= Pn;
        lastRst = (Ln > lastRst) ? Ln : lastRst;
      }
    }
    const float wpack    = rst ? -P : P;              // sign bit = reset flag
    const float cumAfull = (lastRst < 0) ? P : 0.0f;  // full decay incl. zeros

    // A operand: W[M][s], M = lane16; VGPR v of K-block kb holds s = 4kb+v+2hi
    v2f A[4];
#pragma unroll
    for (int kb = 0; kb < 4; ++kb) {
#pragma unroll
      for (int v = 0; v < 2; ++v) {
        const int   s   = 4 * kb + v + 2 * hi;
        const float ws  = __shfl(wpack, s, 32);       // one bpermute: P_s + r_s
        const float bs  = (ws < 0.0f) ? 0.0f : (1.0f - kBeta);
        const float rcp = __builtin_amdgcn_rcpf(fabsf(ws));
        A[kb][v] = (s <= lane16 && lastRst <= s) ? bs * P * rcp : 0.0f;
      }
    }
    // B operand: velocity rows; VGPR v of K-block kb holds row 4kb + 2v + hi
    v2f Bv[4];
#pragma unroll
    for (int e = 0; e < 8; ++e) {
      Bv[e >> 1][e & 1] = p1c[e] - p0c[e];
    }
    // C operand: carry contribution, C[g] = cumA[g + 8*hi] * carry
    v8f acc;
#pragma unroll
    for (int g = 0; g < 8; ++g) {
      acc[g] = __shfl(cumAfull, g + 8 * hi, 32) * carry;
    }
    // momentum tile: four chained 16x16x4 f32 WMMAs
#pragma unroll
    for (int kb = 0; kb < 4; ++kb) {
      acc = __builtin_amdgcn_wmma_f32_16x16x4_f32(
          false, A[kb], false, Bv[kb], (short)0, acc, false, false);
    }
    // epilogue: out = q + momentum (C/D layout: M = g + 8*hi, N = lane16)
    float* orow = out_p + (size_t)t0 * kD;
#pragma unroll
    for (int g = 0; g < 8; ++g) {
      orow[(size_t)g * kD] = qvc[g] + acc[g];
    }
    // next-tile carry: momentum at M=15 sits in acc[7], lanes 16-31
    carry = __shfl(acc[7], 16 + lane16, 32);

    // (5) rotate pipeline registers
#pragma unroll
    for (int e = 0; e < 8; ++e) {
      i1c[e] = i1n[e]; i0c[e] = i0n[e]; qic[e] = qin[e];
      p1c[e] = p1n[e]; p0c[e] = p0n[e]; qvc[e] = qvn[e];
    }
    r_cur = r_next;
  }
}

extern "C" void kernel_launch(void* const* d_in, const int* in_sizes, int n_in,
                              void* d_out, int out_size, void* d_ws, size_t ws_size,
                              hipStream_t stream) {
  const int*   idx        = (const int*)d_in[0];
  const float* reset_mask = (const float*)d_in[1];
  const float* q_emb      = (const float*)d_in[2];
  const float* p_emb      = (const float*)d_in[3];
  float*       out        = (float*)d_out;

  const int waves   = kB * (kD / 16);          // 1024 independent waves
  const int threads = 128;                     // 4 waves per block
  const int blocks  = waves / (threads / 32);  // 256 blocks
  hipLaunchKernelGGL(ema_momentum_wmma, dim3(blocks), dim3(threads), 0, stream,
                     idx, reset_mask, q_emb, p_emb, out);
}